// Layer_63410897158383
// MI455X (gfx1250) — compile-verified
//
#include <hip/hip_runtime.h>
#include <math.h>

#define B_DIM 8
#define S_IN 1024
#define S_OUT 1024
#define D 96
#define G_DIM 8
#define LN_EPS 1e-5f

typedef float v2f __attribute__((ext_vector_type(2)));
typedef float v8f __attribute__((ext_vector_type(8)));
typedef int   v4i_g __attribute__((vector_size(16)));   // matches builtin param pointee

// Async global->LDS (gfx1250 ASYNCcnt path); guarded so absence degrades to
// plain loads instead of a compile failure.
#if defined(__has_builtin)
#  if __has_builtin(__builtin_amdgcn_global_load_async_to_lds_b128)
#    define HAVE_ASYNC_B128 1
#  endif
#endif
#ifndef HAVE_ASYNC_B128
#  define HAVE_ASYNC_B128 0
#endif

// ---------------------------------------------------------------------------
// GEMM1: Z0[r, i] = sum_j x[r, j] * M[i, j]
//   r in [0, 8192), i in [0, 96), K = j in [0, 96)
// One wave per 16(r) x 96(i) strip: 6 WMMA accumulators, the A fragment
// (x row pair) is loaded once and fed to all 6 tiles.
// f32 WMMA layouts (ISA 7.12.2): lane m/n = lane&15; K pair = {2*half, 2*half+1}.
// ---------------------------------------------------------------------------
__global__ __launch_bounds__(32) void gemm1_wmma(const float* __restrict__ x,
                                                 const float* __restrict__ M,
                                                 float* __restrict__ Z0) {
    const int rt   = blockIdx.x;          // 512 row tiles
    const int lane = threadIdx.x;
    const int l15  = lane & 15;
    const int half = lane >> 4;

    const float* xrow = x + (size_t)(rt * 16 + l15) * D;  // A row (m = l15)
    const float* m0   = M + (size_t)l15 * D;              // B[k][n] = M[n*96 + k]

    v8f c[6] = {};
#pragma unroll 2
    for (int k = 0; k < D; k += 4) {
        const int kb = k + 2 * half;
        v2f a;
        a.x = xrow[kb];
        a.y = xrow[kb + 1];
#pragma unroll
        for (int t = 0; t < 6; ++t) {
            const float* mr = m0 + (size_t)(16 * t) * D;
            v2f bb;
            bb.x = mr[kb];
            bb.y = mr[kb + 1];
            c[t] = __builtin_amdgcn_wmma_f32_16x16x4_f32(false, a, false, bb,
                                                         (short)0, c[t], false, false);
        }
    }
    // C/D layout: VGPR v -> row v (lanes 0-15) / row v+8 (lanes 16-31)
    float* zrow = Z0 + (size_t)(rt * 16 + 8 * half) * D + l15;
#pragma unroll
    for (int v = 0; v < 8; ++v)
#pragma unroll
        for (int t = 0; t < 6; ++t)
            zrow[(size_t)v * D + 16 * t] = c[t][v];
}

// ---------------------------------------------------------------------------
// Fused per-s kernel: stage Z0[:,s,:] (8x96) into LDS with async B128 copies
// (ASYNCcnt path), LayerNorm the 8 rows, build W[s] (96x96) in LDS from P and
// cos(2*pi*s/period) -- W never touches HBM -- then
//   V[b,s,i] = Z0[b,s,i] + sum_j LN(Z0)[b,s,j] * W[i,j].
// W row stride 97 (97 mod 64 = 33, odd) -> conflict-free j-loop.
// ---------------------------------------------------------------------------
__global__ __launch_bounds__(256) void phase_mix(const float* __restrict__ Z0,
                                                 const float* __restrict__ P,
                                                 const float* __restrict__ gamma,
                                                 const float* __restrict__ beta,
                                                 float* __restrict__ V) {
    const int s   = blockIdx.x;
    const int tid = threadIdx.x;

    __shared__ __align__(16) float z0[B_DIM][D];
    __shared__ float zl[B_DIM][D];
    __shared__ float w[D * 97];
    __shared__ float mu_s[B_DIM];
    __shared__ float rs_s[B_DIM];

    // ---- Stage the 8 rows Z0[b, s, :] into LDS -----------------------------
#if HAVE_ASYNC_B128
    {   // 8 rows x 24 float4 = 192 b128 transfers, one per lane
        if (tid < B_DIM * (D / 4)) {
            const int bb = tid / (D / 4);
            const int q  = tid % (D / 4);
            const float* g = Z0 + ((size_t)bb * S_IN + s) * D + q * 4;
            __builtin_amdgcn_global_load_async_to_lds_b128(
                (__attribute__((address_space(1))) v4i_g*)g,
                (__attribute__((address_space(3))) v4i_g*)&z0[bb][q * 4],
                0, 0);
        }
#  if defined(__has_builtin) && __has_builtin(__builtin_amdgcn_s_wait_asynccnt)
        __builtin_amdgcn_s_wait_asynccnt(0);
#  else
        asm volatile("s_wait_asynccnt 0x0" ::: "memory");
#  endif
    }
#else
    for (int idx = tid; idx < B_DIM * D; idx += 256) {
        const int bb = idx / D, i = idx % D;
        z0[bb][i] = Z0[((size_t)bb * S_IN + s) * D + i];
    }
#endif
    __syncthreads();

    // ---- Row statistics (8 rows, one thread each; trivial cost) ------------
    if (tid < B_DIM) {
        float mu = 0.f;
        for (int i = 0; i < D; ++i) mu += z0[tid][i];
        mu *= (1.0f / D);
        float var = 0.f;
        for (int i = 0; i < D; ++i) { const float d = z0[tid][i] - mu; var += d * d; }
        var *= (1.0f / D);
        mu_s[tid] = mu;
        rs_s[tid] = rsqrtf(var + LN_EPS);
    }
    __syncthreads();

    for (int idx = tid; idx < B_DIM * D; idx += 256) {
        const int bb = idx / D, i = idx % D;
        zl[bb][i] = (z0[bb][i] - mu_s[bb]) * rs_s[bb] * gamma[i] + beta[i];
    }

    // ---- W[i,j] = sum_g P[i,j,g] * cos(2*pi*s / ((i*96+j)*8 + g + 2)) ------
    const float k2pi = 6.28318530717958647692f * (float)s;
    for (int ij = tid; ij < D * D; ij += 256) {
        float acc = 0.f;
#pragma unroll
        for (int g = 0; g < G_DIM; ++g) {
            const float period = (float)(ij * G_DIM + g + 2);
            acc = __builtin_fmaf(P[ij * G_DIM + g], cosf(k2pi / period), acc);
        }
        w[(ij / D) * 97 + (ij % D)] = acc;
    }
    __syncthreads();

    // ---- V[b,s,i] = Z0[b,s,i] (residual) + sum_j zl[b][j] * W[i][j] --------
    for (int o = tid; o < B_DIM * D; o += 256) {
        const int bb = o / D, i = o % D;
        float acc = z0[bb][i];
        for (int j = 0; j < D; ++j)
            acc = __builtin_fmaf(zl[bb][j], w[i * 97 + j], acc);
        V[((size_t)bb * S_IN + s) * D + i] = acc;
    }
}

// ---------------------------------------------------------------------------
// GEMM2 (1.6 GFLOP, the FLOPs king): out[b,t,i] = sum_s Linker[s,t] * V[b,s,i]
//   A[m=t][k=s] = Linker[s*1024 + t]   (coalesced in t across lanes)
//   B[k=s][n=i] = V[b][s*96 + i]
// One wave per 16(t) x 48(i) strip: A fragment reused across 3 WMMA tiles
// (loads/WMMA = 2.67 vs 4 naive), 1024 waves in flight. K = 1024 -> 768 WMMA
// per wave. Prefetch pulls the next 32-s chunk of both operands toward L2.
// ---------------------------------------------------------------------------
__global__ __launch_bounds__(32) void gemm2_wmma(const float* __restrict__ Linker,
                                                 const float* __restrict__ V,
                                                 float* __restrict__ out) {
    const int ih   = blockIdx.x;          // 0..1  (i half: 48 cols)
    const int tt   = blockIdx.y;          // 0..63 (t tile)
    const int b    = blockIdx.z;          // 0..7
    const int lane = threadIdx.x;
    const int l15  = lane & 15;
    const int half = lane >> 4;

    const float* Acol = Linker + tt * 16 + l15;                        // + s*1024
    const float* Brow = V + (size_t)b * (S_IN * D) + ih * 48 + l15;    // + s*96

    v8f c0 = {}, c1 = {}, c2 = {};
#pragma unroll 2
    for (int s = 0; s < S_IN; s += 4) {
        if ((s & 31) == 0 && s + 32 < S_IN) {
            __builtin_prefetch(Acol + (size_t)(s + 32 + 2 * half) * S_OUT, 0, 1);
            __builtin_prefetch(Brow + (size_t)(s + 32 + 2 * half) * D, 0, 1);
        }
        const int sb = s + 2 * half;
        v2f a;
        a.x = Acol[(size_t)sb * S_OUT];
        a.y = Acol[(size_t)(sb + 1) * S_OUT];
        const float* B0 = Brow + (size_t)sb * D;
        const float* B1 = B0 + D;
        v2f b0, b1, b2;
        b0.x = B0[0];  b0.y = B1[0];
        b1.x = B0[16]; b1.y = B1[16];
        b2.x = B0[32]; b2.y = B1[32];
        c0 = __builtin_amdgcn_wmma_f32_16x16x4_f32(false, a, false, b0, (short)0, c0, false, false);
        c1 = __builtin_amdgcn_wmma_f32_16x16x4_f32(false, a, false, b1, (short)0, c1, false, false);
        c2 = __builtin_amdgcn_wmma_f32_16x16x4_f32(false, a, false, b2, (short)0, c2, false, false);
    }

    const int i0 = ih * 48 + l15;
    float* orow = out + ((size_t)b * S_OUT + tt * 16 + 8 * half) * D + i0;
#pragma unroll
    for (int v = 0; v < 8; ++v) {
        orow[(size_t)v * D + 0]  = c0[v];
        orow[(size_t)v * D + 16] = c1[v];
        orow[(size_t)v * D + 32] = c2[v];
    }
}

// ---------------------------------------------------------------------------
extern "C" void kernel_launch(void* const* d_in, const int* in_sizes, int n_in,
                              void* d_out, int out_size, void* d_ws, size_t ws_size,
                              hipStream_t stream) {
    const float* x      = (const float*)d_in[0];
    const float* M      = (const float*)d_in[1];
    const float* P      = (const float*)d_in[2];
    const float* Linker = (const float*)d_in[3];
    const float* gamma  = (const float*)d_in[4];
    const float* beta   = (const float*)d_in[5];
    float* out = (float*)d_out;

    float* Z0 = (float*)d_ws;                             // 8192*96 f32 = 3.15 MB
    float* V  = Z0 + (size_t)B_DIM * S_IN * D;            // 8192*96 f32 = 3.15 MB

    // 1) Z0 = x @ M^T   (WMMA f32; 512 waves, 16x96 strip each)
    gemm1_wmma<<<dim3(512), dim3(32), 0, stream>>>(x, M, Z0);

    // 2) Fused LN + W(s)-in-LDS + T + residual-merge -> V = T + Z0
    phase_mix<<<dim3(S_IN), dim3(256), 0, stream>>>(Z0, P, gamma, beta, V);

    // 3) out[b] = Linker^T @ V[b]  (WMMA f32; 1024 waves, 16x48 strip, K=1024)
    gemm2_wmma<<<dim3(2, 64, B_DIM), dim3(32), 0, stream>>>(Linker, V, out);
}